// CombinedModel_43593918054899
// MI455X (gfx1250) — compile-verified
//
#include <hip/hip_runtime.h>
#include <hip/hip_bf16.h>
#include <math.h>

typedef __attribute__((ext_vector_type(2))) float v2f;
typedef __attribute__((ext_vector_type(8))) float v8f;

#define LN_EPS    1e-5f
#define SM_EPS    1e-16f
#define NEG_SLOPE 0.2f
#define O_CH      32
#define RREL      15

// ---------------- utility ----------------

__device__ __forceinline__ void atomicMaxF(float* addr, float v) {
    int iv = __float_as_int(v);
    if (iv >= 0) atomicMax((int*)addr, iv);
    else         atomicMin((unsigned int*)addr, (unsigned int)iv);
}

__device__ __forceinline__ float siluf(float y) { return y / (1.0f + expf(-y)); }

__global__ void k_fill(float* __restrict__ p, float v, long n) {
    long i = (long)blockIdx.x * blockDim.x + threadIdx.x;
    if (i < n) p[i] = v;
}

// ---------------- pos normalization ----------------

__global__ void k_pos_mean(const float* __restrict__ db, int N, float* __restrict__ scal) {
    __shared__ float s0, s1, s2;
    if (threadIdx.x == 0) { s0 = 0.f; s1 = 0.f; s2 = 0.f; }
    __syncthreads();
    float a0 = 0.f, a1 = 0.f, a2 = 0.f;
    for (long i = (long)blockIdx.x * blockDim.x + threadIdx.x; i < N;
         i += (long)gridDim.x * blockDim.x) {
        const float* r = db + i * 35;
        a0 += r[0]; a1 += r[1]; a2 += r[2];
    }
    atomicAdd(&s0, a0); atomicAdd(&s1, a1); atomicAdd(&s2, a2);
    __syncthreads();
    if (threadIdx.x == 0) {
        atomicAdd(&scal[0], s0); atomicAdd(&scal[1], s1); atomicAdd(&scal[2], s2);
    }
}

__global__ void k_pos_absmax(const float* __restrict__ db, int N, float* __restrict__ scal) {
    __shared__ int sm;
    if (threadIdx.x == 0) sm = 0;
    __syncthreads();
    float invN = 1.0f / (float)N;
    float mu0 = scal[0] * invN, mu1 = scal[1] * invN, mu2 = scal[2] * invN;
    float m = 0.f;
    for (long i = (long)blockIdx.x * blockDim.x + threadIdx.x; i < N;
         i += (long)gridDim.x * blockDim.x) {
        const float* r = db + i * 35;
        m = fmaxf(m, fabsf(r[0] - mu0));
        m = fmaxf(m, fabsf(r[1] - mu1));
        m = fmaxf(m, fabsf(r[2] - mu2));
    }
    atomicMax(&sm, __float_as_int(m));   // nonneg floats: int order == float order
    __syncthreads();
    if (threadIdx.x == 0) atomicMax((int*)&scal[3], sm);
}

__global__ void k_pos_cov(const float* __restrict__ db, int N, float* __restrict__ scal) {
    __shared__ float s[6];
    if (threadIdx.x < 6) s[threadIdx.x] = 0.f;
    __syncthreads();
    float invN = 1.0f / (float)N;
    float mu0 = scal[0] * invN, mu1 = scal[1] * invN, mu2 = scal[2] * invN;
    float sc = 0.999999f / scal[3];
    float c00 = 0, c01 = 0, c02 = 0, c11 = 0, c12 = 0, c22 = 0;
    for (long i = (long)blockIdx.x * blockDim.x + threadIdx.x; i < N;
         i += (long)gridDim.x * blockDim.x) {
        const float* r = db + i * 35;
        float p0 = (r[0] - mu0) * sc, p1 = (r[1] - mu1) * sc, p2 = (r[2] - mu2) * sc;
        c00 += p0 * p0; c01 += p0 * p1; c02 += p0 * p2;
        c11 += p1 * p1; c12 += p1 * p2; c22 += p2 * p2;
    }
    atomicAdd(&s[0], c00); atomicAdd(&s[1], c01); atomicAdd(&s[2], c02);
    atomicAdd(&s[3], c11); atomicAdd(&s[4], c12); atomicAdd(&s[5], c22);
    __syncthreads();
    if (threadIdx.x < 6) atomicAdd(&scal[4 + threadIdx.x], s[threadIdx.x]);
}

// 3x3 symmetric Jacobi eigensolver; eigenvectors sorted by ascending eigenvalue.
__global__ void k_eig3(float* __restrict__ scal) {
    if (threadIdx.x != 0 || blockIdx.x != 0) return;
    double A[3][3] = {{scal[4], scal[5], scal[6]},
                      {scal[5], scal[7], scal[8]},
                      {scal[6], scal[8], scal[9]}};
    double V[3][3] = {{1, 0, 0}, {0, 1, 0}, {0, 0, 1}};
    for (int sweep = 0; sweep < 30; ++sweep) {
        double off = fabs(A[0][1]) + fabs(A[0][2]) + fabs(A[1][2]);
        if (off < 1e-18) break;
        for (int p = 0; p < 2; ++p)
            for (int q = p + 1; q < 3; ++q) {
                double apq = A[p][q];
                if (fabs(apq) < 1e-22) continue;
                double theta = (A[q][q] - A[p][p]) / (2.0 * apq);
                double t = (theta >= 0 ? 1.0 : -1.0) /
                           (fabs(theta) + sqrt(theta * theta + 1.0));
                double c = 1.0 / sqrt(t * t + 1.0), s = t * c;
                for (int k = 0; k < 3; ++k) {
                    double akp = A[k][p], akq = A[k][q];
                    A[k][p] = c * akp - s * akq;
                    A[k][q] = s * akp + c * akq;
                }
                for (int k = 0; k < 3; ++k) {
                    double apk = A[p][k], aqk = A[q][k];
                    A[p][k] = c * apk - s * aqk;
                    A[q][k] = s * apk + c * aqk;
                }
                for (int k = 0; k < 3; ++k) {
                    double vkp = V[k][p], vkq = V[k][q];
                    V[k][p] = c * vkp - s * vkq;
                    V[k][q] = s * vkp + c * vkq;
                }
            }
    }
    int idx[3] = {0, 1, 2};
    for (int i = 0; i < 2; ++i)
        for (int j = i + 1; j < 3; ++j)
            if (A[idx[j]][idx[j]] < A[idx[i]][idx[i]]) {
                int t = idx[i]; idx[i] = idx[j]; idx[j] = t;
            }
    for (int k = 0; k < 3; ++k)
        for (int c = 0; c < 3; ++c)
            scal[16 + k * 3 + c] = (float)V[k][idx[c]];
}

__global__ void k_feat_stats(const float* __restrict__ db, int N, float* __restrict__ scal) {
    __shared__ float s1, s2;
    if (threadIdx.x == 0) { s1 = 0.f; s2 = 0.f; }
    __syncthreads();
    float a = 0.f, b = 0.f;
    long n32 = (long)N * 32;
    for (long i = (long)blockIdx.x * blockDim.x + threadIdx.x; i < n32;
         i += (long)gridDim.x * blockDim.x) {
        long n = i >> 5; int c = (int)(i & 31);
        float v = db[n * 35 + 3 + c];
        a += v; b += v * v;
    }
    atomicAdd(&s1, a); atomicAdd(&s2, b);
    __syncthreads();
    if (threadIdx.x == 0) { atomicAdd(&scal[10], s1); atomicAdd(&scal[11], s2); }
}

// writes node features with row stride 36 (K padded to multiple of 4; col 35 = 0)
__global__ void k_build_x(const float* __restrict__ db, const float* __restrict__ scal,
                          const float* __restrict__ lnw, const float* __restrict__ lnb,
                          float* __restrict__ x, int N) {
    int n = blockIdx.x * blockDim.x + threadIdx.x;
    if (n >= N) return;
    float invN = 1.0f / (float)N;
    float mu0 = scal[0] * invN, mu1 = scal[1] * invN, mu2 = scal[2] * invN;
    float sc = 0.999999f / scal[3];
    const float* r = db + (long)n * 35;
    float p0 = (r[0] - mu0) * sc, p1 = (r[1] - mu1) * sc, p2 = (r[2] - mu2) * sc;
    float* xo = x + (long)n * 36;
#pragma unroll
    for (int c = 0; c < 3; ++c)
        xo[c] = p0 * scal[16 + 0 + c] + p1 * scal[16 + 3 + c] + p2 * scal[16 + 6 + c];
    float cnt = (float)N * 32.0f;
    float mu = scal[10] / cnt;
    float var = fmaxf(scal[11] / cnt - mu * mu, 0.f);
    float inv = 1.0f / (sqrtf(var) + LN_EPS);
#pragma unroll
    for (int c = 0; c < 32; ++c)
        xo[3 + c] = (r[3 + c] - mu) * inv * lnw[c] + lnb[c];
    xo[35] = 0.0f;                           // K padding
}

// transpose + zero-pad the relation weights:  Wt[r][col][kk] , kk in [0,KP)
__global__ void k_prepW(const float* __restrict__ W, float* __restrict__ Wt,
                        int K, int KP) {
    int i = blockIdx.x * blockDim.x + threadIdx.x;
    if (i >= RREL * O_CH * KP) return;
    int kk = i % KP;
    int rest = i / KP;
    int col = rest & 31;
    int r = rest >> 5;
    Wt[i] = (kk < K) ? W[((long)r * K + kk) * O_CH + col] : 0.0f;
}

// ---------------- per-relation GEMM via f32 WMMA ----------------
// XR[r] = X(N x KP, padded) @ Wt[r]^T.  One wave computes a 16x16 output tile
// with a fully unrolled chain of V_WMMA_F32_16X16X4_F32; all loads are
// unconditional global_load_b64 (A row-major, B transposed so K is contiguous).

template <int KP>
__global__ void k_rgemm_wmma(const float* __restrict__ X, const float* __restrict__ Wt,
                             float* __restrict__ XR, int N) {
    int wave = (int)(((long)blockIdx.x * blockDim.x + threadIdx.x) >> 5);
    int tilesPerRel = (N >> 4) * 2;                 // 16-row blocks x 2 col-halves
    if (wave >= RREL * tilesPerRel) return;         // whole-wave exit: EXEC all-1s
    int r   = wave / tilesPerRel;
    int rem = wave - r * tilesPerRel;
    int m0  = (rem >> 1) << 4;
    int c0  = (rem & 1) << 4;
    int lane = threadIdx.x & 31;
    int half = lane >> 4;                           // lanes 16-31 carry K+2,K+3
    int lm   = lane & 15;
    const float* Arow = X + (long)(m0 + lm) * KP + half * 2;
    const float* Brow = Wt + ((long)r * O_CH + c0 + lm) * KP + half * 2;
    v8f acc = {0.f, 0.f, 0.f, 0.f, 0.f, 0.f, 0.f, 0.f};
#pragma unroll
    for (int k = 0; k < KP; k += 4) {
        v2f a = *(const v2f*)(Arow + k);
        v2f b = *(const v2f*)(Brow + k);
        acc = __builtin_amdgcn_wmma_f32_16x16x4_f32(false, a, false, b,
                                                    (short)0, acc, false, false);
    }
    float* out = XR + ((long)r * N + m0) * O_CH + c0 + lm;
#pragma unroll
    for (int g = 0; g < 8; ++g) {
        int row = g + half * 8;                     // C layout: vgpr g = M=g / M=g+8
        out[(long)row * O_CH] = acc[g];
    }
}

// per-(relation,node) attention scalars: Q = xr.q, K = xr.k  (kills edge gather traffic)
__global__ void k_qk(const float* __restrict__ xr, const float* __restrict__ q,
                     const float* __restrict__ k, float* __restrict__ Qv,
                     float* __restrict__ Kv, long RN) {
    long i = (long)blockIdx.x * blockDim.x + threadIdx.x;
    if (i >= RN) return;
    const float4* row = (const float4*)(xr + (i << 5));
    const float4* qv4 = (const float4*)q;
    const float4* kv4 = (const float4*)k;
    float aq = 0.f, ak = 0.f;
#pragma unroll
    for (int c = 0; c < 8; ++c) {
        float4 xv = row[c], qv = qv4[c], kv = kv4[c];
        aq += xv.x * qv.x + xv.y * qv.y + xv.z * qv.z + xv.w * qv.w;
        ak += xv.x * kv.x + xv.y * kv.y + xv.z * kv.z + xv.w * kv.w;
    }
    Qv[i] = aq;
    Kv[i] = ak;
}

// ---------------- edge phase: logits -> segment softmax -> scatter ----------------

__global__ void k_edge_alpha(const float* __restrict__ Qv, const float* __restrict__ Kv,
                             const int* __restrict__ ei, const int* __restrict__ et,
                             float* __restrict__ alpha, float* __restrict__ amax,
                             long E, int N) {
    long e = (long)blockIdx.x * blockDim.x + threadIdx.x;
    if (e >= E) return;
    int s = ei[e], d = ei[E + e], r = et[e];
    float a = Qv[(long)r * N + d] + Kv[(long)r * N + s];
    a = (a > 0.f) ? a : NEG_SLOPE * a;
    alpha[e] = a;
    atomicMaxF(&amax[d], a);
}

__global__ void k_edge_exp(float* __restrict__ alpha, const float* __restrict__ amax,
                           float* __restrict__ den, const int* __restrict__ ei, long E) {
    long e = (long)blockIdx.x * blockDim.x + threadIdx.x;
    if (e >= E) return;
    int d = ei[E + e];
    float ex = expf(alpha[e] - amax[d]);
    alpha[e] = ex;
    atomicAdd(&den[d], ex);
}

// 8 threads per edge, 4 channels each; xr rows are L2-resident (192MB table == L2)
__global__ void k_edge_scatter(const float* __restrict__ alpha, const float* __restrict__ den,
                               const int* __restrict__ ei, const int* __restrict__ et,
                               const float* __restrict__ xr, float* __restrict__ h,
                               long E, int N) {
    long t = (long)blockIdx.x * blockDim.x + threadIdx.x;
    long e = t >> 3;
    if (e >= E) return;
    int g = (int)(t & 7);
    int s = ei[e], d = ei[E + e], r = et[e];
    float coef = alpha[e] / (den[d] + SM_EPS);
    const float4* row = (const float4*)(xr + (((long)r * N + s) << 5));
    float4 v = row[g];
    float* hd = h + ((long)d << 5) + (g << 2);
    atomicAdd(hd + 0, coef * v.x);
    atomicAdd(hd + 1, coef * v.y);
    atomicAdd(hd + 2, coef * v.z);
    atomicAdd(hd + 3, coef * v.w);
}

// ---------------- bias + residual, graph-LN + SiLU ----------------

__global__ void k_bias_res(float* __restrict__ h, const float* __restrict__ bias,
                           const float* __restrict__ prev, long n32) {
    long i = (long)blockIdx.x * blockDim.x + threadIdx.x;
    if (i >= n32) return;
    float v = h[i] + bias[(int)(i & 31)];
    if (prev) v += prev[i];
    h[i] = v;
}

__global__ void k_stats(const float* __restrict__ p, long n, float* __restrict__ s1,
                        float* __restrict__ s2) {
    __shared__ float sh1, sh2;
    if (threadIdx.x == 0) { sh1 = 0.f; sh2 = 0.f; }
    __syncthreads();
    float a = 0.f, b = 0.f;
    for (long i = (long)blockIdx.x * blockDim.x + threadIdx.x; i < n;
         i += (long)gridDim.x * blockDim.x) {
        float v = p[i];
        a += v; b += v * v;
    }
    atomicAdd(&sh1, a); atomicAdd(&sh2, b);
    __syncthreads();
    if (threadIdx.x == 0) { atomicAdd(s1, sh1); atomicAdd(s2, sh2); }
}

__global__ void k_ln_silu(const float* __restrict__ h, const float* __restrict__ scal,
                          const float* __restrict__ lnw, const float* __restrict__ lnb,
                          float* __restrict__ cur, long n32) {
    long i = (long)blockIdx.x * blockDim.x + threadIdx.x;
    if (i >= n32) return;
    float cnt = (float)n32;
    float mu = scal[12] / cnt;
    float var = fmaxf(scal[13] / cnt - mu * mu, 0.f);
    float inv = 1.0f / (sqrtf(var) + LN_EPS);
    int c = (int)(i & 31);
    float y = (h[i] - mu) * inv * lnw[c] + lnb[c];
    cur[i] = siluf(y);
}

// ---------------- pooling + head ----------------

__global__ void k_pool(const float* __restrict__ cur, long n32, float* __restrict__ pool) {
    __shared__ float sp[32];
    if (threadIdx.x < 32) sp[threadIdx.x] = 0.f;
    __syncthreads();
    for (long i = (long)blockIdx.x * blockDim.x + threadIdx.x; i < n32;
         i += (long)gridDim.x * blockDim.x)
        atomicAdd(&sp[(int)(i & 31)], cur[i]);
    __syncthreads();
    if (threadIdx.x < 32) atomicAdd(&pool[threadIdx.x], sp[threadIdx.x]);
}

__global__ void k_final(const float* __restrict__ pool, const float* __restrict__ W,
                        const float* __restrict__ b, float* __restrict__ out, int N) {
    int j = blockIdx.x * blockDim.x + threadIdx.x;
    if (j >= 256) return;
    float invN = 1.0f / (float)N;
    float acc = b[j];
#pragma unroll
    for (int c = 0; c < 32; ++c)
        acc += (pool[c] * invN) * W[j * 32 + c];
    out[j] = siluf(acc);
}

// ---------------- driver ----------------

static inline int gb(long n, int b) { return (int)((n + b - 1) / b); }

extern "C" void kernel_launch(void* const* d_in, const int* in_sizes, int n_in,
                              void* d_out, int out_size, void* d_ws, size_t ws_size,
                              hipStream_t stream) {
    (void)n_in; (void)out_size; (void)ws_size;
    const float* db1  = (const float*)d_in[0];
    const int*   ei   = (const int*)d_in[2];
    const int*   et   = (const int*)d_in[3];
    const float* W0   = (const float*)d_in[6];
    const float* q0   = (const float*)d_in[7];
    const float* k0   = (const float*)d_in[8];
    const float* b0   = (const float*)d_in[9];
    const float* Wsin = (const float*)d_in[10];
    const float* qs   = (const float*)d_in[11];
    const float* ks   = (const float*)d_in[12];
    const float* bs   = (const float*)d_in[13];
    const float* lnw  = (const float*)d_in[14];
    const float* lnb  = (const float*)d_in[15];
    const float* ln1w = (const float*)d_in[16];
    const float* ln1b = (const float*)d_in[17];
    const float* l1W  = (const float*)d_in[18];
    const float* l1b  = (const float*)d_in[19];
    float* out = (float*)d_out;
    int  N = in_sizes[0] / 35;
    long E = in_sizes[3];

    // workspace carve-out (256B aligned)
    char* wsp = (char*)d_ws;
    auto alloc = [&](size_t bytes) -> float* {
        float* p = (float*)wsp;
        wsp += (bytes + 255) & ~((size_t)255);
        return p;
    };
    float* x36  = alloc((size_t)N * 36 * 4);            // padded layer-0 input
    float* xr   = alloc((size_t)RREL * N * O_CH * 4);   // 192 MB == MI455X L2
    float* Wt   = alloc((size_t)RREL * O_CH * 36 * 4);  // transposed padded weights
    float* Qv   = alloc((size_t)RREL * N * 4);
    float* Kv   = alloc((size_t)RREL * N * 4);
    float* alp  = alloc((size_t)E * 4);
    float* amax = alloc((size_t)N * 4);
    float* den  = alloc((size_t)N * 4);
    float* h    = alloc((size_t)N * O_CH * 4);
    float* cur  = alloc((size_t)N * O_CH * 4);
    float* scal = alloc(128 * 4);

    const int B = 256;
    const long n32 = (long)N * O_CH;

    // ---- preprocessing: pos normalization + feature graph-LN ----
    k_fill<<<1, 128, 0, stream>>>(scal, 0.f, 128);
    k_pos_mean  <<<512, B, 0, stream>>>(db1, N, scal);
    k_pos_absmax<<<512, B, 0, stream>>>(db1, N, scal);
    k_pos_cov   <<<512, B, 0, stream>>>(db1, N, scal);
    k_eig3      <<<1, 1, 0, stream>>>(scal);
    k_feat_stats<<<512, B, 0, stream>>>(db1, N, scal);
    k_build_x<<<gb(N, B), B, 0, stream>>>(db1, scal, ln1w, ln1b, x36, N);

    // ---- 4 RGAT layers ----
    for (int i = 0; i < 4; ++i) {
        int          K  = (i == 0) ? 35 : 32;
        int          KP = (i == 0) ? 36 : 32;
        const float* Wl = (i == 0) ? W0 : Wsin + (size_t)(i - 1) * RREL * O_CH * O_CH;
        const float* ql = (i == 0) ? q0 : qs + (size_t)(i - 1) * O_CH;
        const float* kl = (i == 0) ? k0 : ks + (size_t)(i - 1) * O_CH;
        const float* bl = (i == 0) ? b0 : bs + (size_t)(i - 1) * O_CH;

        k_prepW<<<gb((long)RREL * O_CH * KP, B), B, 0, stream>>>(Wl, Wt, K, KP);

        long waves = (long)RREL * (N >> 4) * 2;
        if (i == 0)
            k_rgemm_wmma<36><<<gb(waves * 32, B), B, 0, stream>>>(x36, Wt, xr, N);
        else
            k_rgemm_wmma<32><<<gb(waves * 32, B), B, 0, stream>>>(cur, Wt, xr, N);

        k_qk<<<gb((long)RREL * N, B), B, 0, stream>>>(xr, ql, kl, Qv, Kv, (long)RREL * N);

        k_fill<<<gb(N, B), B, 0, stream>>>(amax, -INFINITY, N);
        k_fill<<<gb(N, B), B, 0, stream>>>(den, 0.f, N);
        k_fill<<<gb(n32, B), B, 0, stream>>>(h, 0.f, n32);

        k_edge_alpha  <<<gb(E, B), B, 0, stream>>>(Qv, Kv, ei, et, alp, amax, E, N);
        k_edge_exp    <<<gb(E, B), B, 0, stream>>>(alp, amax, den, ei, E);
        k_edge_scatter<<<gb(E * 8, B), B, 0, stream>>>(alp, den, ei, et, xr, h, E, N);

        k_bias_res<<<gb(n32, B), B, 0, stream>>>(h, bl, (i == 0) ? nullptr : cur, n32);
        k_fill<<<1, 64, 0, stream>>>(scal + 12, 0.f, 2);
        k_stats<<<512, B, 0, stream>>>(h, n32, &scal[12], &scal[13]);
        k_ln_silu<<<gb(n32, B), B, 0, stream>>>(h, scal, lnw + (size_t)i * O_CH,
                                                lnb + (size_t)i * O_CH, cur, n32);
    }

    // ---- mean pool + head ----
    k_fill<<<1, 64, 0, stream>>>(scal + 32, 0.f, 32);
    k_pool<<<512, B, 0, stream>>>(cur, n32, scal + 32);
    k_final<<<1, 256, 0, stream>>>(scal + 32, l1W, l1b, out, N);
}